// HybridQuantumClassifier_65481071400075
// MI455X (gfx1250) — compile-verified
//
#include <hip/hip_runtime.h>
#include <hip/hip_bf16.h>

#define N_WIRES 10
#define QDIM 1024          // 2^10 amplitudes
#define DEPTH 4
#define BLK 256

typedef __attribute__((ext_vector_type(2))) float v2f;
typedef __attribute__((ext_vector_type(8))) float v8f;

// ---------------------------------------------------------------------------
// Kernel 1: one workgroup per sample; statevector lives in LDS (8KB).
// Emits per-sample z[10] (un-normalized sum weights handled later).
// ---------------------------------------------------------------------------
__global__ __launch_bounds__(BLK) void qsim_kernel(const float* __restrict__ x,      // (2048, 10)
                                                   const float* __restrict__ params, // (4, 10)
                                                   float* __restrict__ zout)         // (2048, 10)
{
    __shared__ float sre[QDIM];
    __shared__ float sim[QDIM];
    __shared__ float red[8 * N_WIRES];

    const int tid = threadIdx.x;
    const int smp = blockIdx.x;

    // |0...0> init
#pragma unroll
    for (int q = 0; q < 4; ++q) {
        int i = tid + BLK * q;
        sre[i] = (i == 0) ? 1.0f : 0.0f;
        sim[i] = 0.0f;
    }
    __syncthreads();

    // ---- RX(x[w]) on each wire: U = [[c, -i s], [-i s, c]] ----
    for (int w = 0; w < N_WIRES; ++w) {
        const float half = 0.5f * x[smp * N_WIRES + w];
        const float c  = __cosf(half);
        const float sn = __sinf(half);
        const int shift = 9 - w;          // wire w <-> bit (9-w)
        const int dr = 1 << shift;
#pragma unroll
        for (int q = 0; q < 2; ++q) {
            int p  = tid + BLK * q;       // pair index 0..511
            int l  = p >> shift;
            int r  = p & (dr - 1);
            int i0 = (l << (shift + 1)) + r;
            int i1 = i0 + dr;
            float a0r = sre[i0], a0i = sim[i0];
            float a1r = sre[i1], a1i = sim[i1];
            sre[i0] = c * a0r + sn * a1i;   // c*a0 + (-i s)*a1
            sim[i0] = c * a0i - sn * a1r;
            sre[i1] = c * a1r + sn * a0i;   // (-i s)*a0 + c*a1
            sim[i1] = c * a1i - sn * a0r;
        }
        __syncthreads();
    }

    // ---- DEPTH layers: RY(params) per wire, then CNOT ladder ----
    for (int layer = 0; layer < DEPTH; ++layer) {
        for (int w = 0; w < N_WIRES; ++w) {
            const float ph = 0.5f * params[layer * N_WIRES + w];
            const float cp = __cosf(ph);
            const float sp = __sinf(ph);
            const int shift = 9 - w;
            const int dr = 1 << shift;
#pragma unroll
            for (int q = 0; q < 2; ++q) {
                int p  = tid + BLK * q;
                int l  = p >> shift;
                int r  = p & (dr - 1);
                int i0 = (l << (shift + 1)) + r;
                int i1 = i0 + dr;
                float a0r = sre[i0], a0i = sim[i0];
                float a1r = sre[i1], a1i = sim[i1];
                sre[i0] = cp * a0r - sp * a1r;   // real 2x2 rotation
                sim[i0] = cp * a0i - sp * a1i;
                sre[i1] = sp * a0r + cp * a1r;
                sim[i1] = sp * a0i + cp * a1i;
            }
            __syncthreads();
        }
        // CNOT(0,1)...CNOT(8,9) composed == Gray-code gather: new[j] = old[j ^ (j>>1)]
        float tr[4], ti[4];
#pragma unroll
        for (int q = 0; q < 4; ++q) {
            int j   = tid + BLK * q;
            int src = j ^ (j >> 1);
            tr[q] = sre[src];
            ti[q] = sim[src];
        }
        __syncthreads();
#pragma unroll
        for (int q = 0; q < 4; ++q) {
            int j = tid + BLK * q;
            sre[j] = tr[q];
            sim[j] = ti[q];
        }
        __syncthreads();
    }

    // ---- z[w] = sum_i p_i * (+1 if bit(9-w)==0 else -1) ----
    float zl[N_WIRES];
#pragma unroll
    for (int w = 0; w < N_WIRES; ++w) zl[w] = 0.0f;
#pragma unroll
    for (int q = 0; q < 4; ++q) {
        int i = tid + BLK * q;
        float p = sre[i] * sre[i] + sim[i] * sim[i];
#pragma unroll
        for (int w = 0; w < N_WIRES; ++w)
            zl[w] += ((i >> (9 - w)) & 1) ? -p : p;
    }
    // wave32 butterfly reduce
#pragma unroll
    for (int off = 16; off > 0; off >>= 1)
#pragma unroll
        for (int w = 0; w < N_WIRES; ++w)
            zl[w] += __shfl_down(zl[w], off, 32);

    const int wave = tid >> 5;
    const int lane = tid & 31;
    if (lane == 0)
#pragma unroll
        for (int w = 0; w < N_WIRES; ++w) red[wave * N_WIRES + w] = zl[w];
    __syncthreads();
    if (tid < N_WIRES) {
        float acc = 0.0f;
#pragma unroll
        for (int v = 0; v < 8; ++v) acc += red[v * N_WIRES + tid];
        zout[smp * N_WIRES + tid] = acc;
    }
}

// ---------------------------------------------------------------------------
// Kernel 2: mean over the 128-step time axis, then the 16x10 @ 10x2 head
// done with V_WMMA_F32_16X16X4_F32 (K padded 10 -> 12, 3 chained WMMAs).
// ---------------------------------------------------------------------------
__global__ __launch_bounds__(BLK) void head_kernel(const float* __restrict__ z,      // (2048, 10)
                                                   const float* __restrict__ head_w, // (2, 10)
                                                   const float* __restrict__ head_b, // (2,)
                                                   float* __restrict__ out)          // (16, 2)
{
    __shared__ float Am[16][12];   // z_mean, K-padded
    __shared__ float Bm[12][16];   // head_w^T, N-padded

    const int tid = threadIdx.x;

    if (tid < 192) {
        Am[tid / 12][tid % 12] = 0.0f;
        Bm[tid / 16][tid % 16] = 0.0f;
    }
    __syncthreads();

    if (tid < 160) {                       // 16 batch rows x 10 wires
        int b = tid / N_WIRES, w = tid % N_WIRES;
        const float* p = z + (size_t)b * 128 * N_WIRES + w;
        float acc = 0.0f;
        for (int t = 0; t < 128; ++t) acc += p[t * N_WIRES];
        Am[b][w] = acc * (1.0f / 128.0f);
    } else if (tid < 180) {                // B[k][n] = head_w[n][k]
        int t2 = tid - 160;
        int n = t2 / N_WIRES, k = t2 % N_WIRES;
        Bm[k][n] = head_w[n * N_WIRES + k];
    }
    __syncthreads();

    if (tid < 32) {                        // wave 0, EXEC all ones
        const int lane = tid;
        const int m  = lane & 15;          // A: M index, B: N index
        const int hi = lane >> 4;          // half-wave selects K pair {0,1} vs {2,3}
        v8f c = {};
#if defined(__has_builtin) && __has_builtin(__builtin_amdgcn_wmma_f32_16x16x4_f32)
#pragma unroll
        for (int k0 = 0; k0 < 12; k0 += 4) {
            v2f a, b;
            a.x = Am[m][k0 + (hi ? 2 : 0)];
            a.y = Am[m][k0 + (hi ? 3 : 1)];
            b.x = Bm[k0 + (hi ? 2 : 0)][m];
            b.y = Bm[k0 + (hi ? 3 : 1)][m];
            c = __builtin_amdgcn_wmma_f32_16x16x4_f32(false, a, false, b,
                                                      (short)0, c, false, false);
        }
        // D layout: VGPR r -> row (r + hi*8), col = lane&15
        if (m < 2) {
#pragma unroll
            for (int r = 0; r < 8; ++r)
                out[(r + hi * 8) * 2 + m] = c[r] + head_b[m];
        }
#else
        // scalar fallback (should not trigger on gfx1250)
        if (lane < 16) {
            float o0 = head_b[0], o1 = head_b[1];
            for (int k = 0; k < N_WIRES; ++k) {
                o0 += Am[lane][k] * Bm[k][0];
                o1 += Am[lane][k] * Bm[k][1];
            }
            out[lane * 2 + 0] = o0;
            out[lane * 2 + 1] = o1;
        }
        (void)c;
#endif
    }
}

// ---------------------------------------------------------------------------
extern "C" void kernel_launch(void* const* d_in, const int* in_sizes, int n_in,
                              void* d_out, int out_size, void* d_ws, size_t ws_size,
                              hipStream_t stream) {
    const float* x      = (const float*)d_in[0];  // (16,128,10) = 2048 samples
    const float* params = (const float*)d_in[1];  // (4,10)
    const float* head_w = (const float*)d_in[2];  // (2,10)
    const float* head_b = (const float*)d_in[3];  // (2,)
    float* out  = (float*)d_out;                  // (16,2)
    float* zbuf = (float*)d_ws;                   // 2048*10 floats = 80KB scratch

    qsim_kernel<<<2048, BLK, 0, stream>>>(x, params, zbuf);
    head_kernel<<<1, BLK, 0, stream>>>(zbuf, head_w, head_b, out);
}